// EdgeNetwork_10222022164946
// MI455X (gfx1250) — compile-verified
//
#include <hip/hip_runtime.h>

typedef float v2f __attribute__((ext_vector_type(2)));
typedef float v4f __attribute__((ext_vector_type(4)));
typedef float v8f __attribute__((ext_vector_type(8)));

#define ATOM_DIM 16
#define NCHUNK   68   // 272 / 4 : 64 kernel K-chunks + 4 bias K-chunks
#define TPW      4    // edge-tiles (of 16 edges) per wave

// ---------------------------------------------------------------------------
// Zero-fill output (harness poisons d_out; we accumulate with atomics).
// ---------------------------------------------------------------------------
__global__ void zero_kernel(float* __restrict__ out, int n) {
    int i = blockIdx.x * blockDim.x + threadIdx.x;
    if (i < n) out[i] = 0.0f;
}

// ---------------------------------------------------------------------------
// Fused EdgeNetwork as ONE 16x272x16 GEMM per 16-edge tile:
//
//   msgs = U @ W,   U[e][K] = bf[e][K>>4] * neigh[e][K&15]   (K = 0..255)
//                   U[e][256+j] = neigh[e][j]                 (bias columns)
//                   W[K][i]  = kernel[K>>4][i*16 + (K&15)]    (K < 256)
//                   W[256+j][i] = bias[i*16 + j]
//
// Each wave processes TPW=4 tiles simultaneously so every B-operand LDS read
// feeds 4 WMMAs: LDS bandwidth demand drops 4x, making the loop bound by the
// matrix pipe instead of the (WGP-shared) LDS pipe. The 4 per-tile
// accumulators are 4 independent WMMA dependency chains.
//
// WMMA f32 layouts (wave32):
//   A 16x4 : v2f per lane; element (M = lane%16, K = 2*(lane/16)+comp)
//   B 4x16 : v2f per lane; element (K = 2*(lane/16)+comp, N = lane%16)
//   C/D    : v8f per lane; element (M = v + 8*(lane/16), N = lane%16)
//
// ldsW[c0][half][n][jj] = W[4*c0 + 2*half + jj][n]  -> one conflict-free
// ds_load_b64 per lane per chunk (compiler pairs them into ds_load_2addr).
// ---------------------------------------------------------------------------
__global__ __launch_bounds__(256) void edge_net_kernel(
    const float* __restrict__ atom,   // N_ATOMS x 16
    const float* __restrict__ bond,   // N_EDGES x 16
    const int*   __restrict__ pair,   // N_EDGES x 2  ([e][0]=seg, [e][1]=nbr)
    const float* __restrict__ kern,   // 16 x 256
    const float* __restrict__ bias,   // 256
    float*       __restrict__ out,    // N_ATOMS x 16 (pre-zeroed)
    int nEdges)
{
    __shared__ float ldsW[NCHUNK * 2 * 16 * 2];   // 17408 B

    // Stage kernel+bias into permuted flat-K layout (once per block).
    for (int f = threadIdx.x; f < NCHUNK * 64; f += blockDim.x) {
        int jj   = f & 1;
        int n    = (f >> 1) & 15;
        int half = (f >> 5) & 1;
        int c0   = f >> 6;
        int K    = 4 * c0 + 2 * half + jj;
        int col  = n * 16 + (K & 15);
        ldsW[f]  = (K < 256) ? kern[(K >> 4) * 256 + col] : bias[col];
    }
    __syncthreads();

    const int lane = threadIdx.x & 31;
    const int wave = threadIdx.x >> 5;
    const int e16  = lane & 15;
    const int half = lane >> 4;

    const int job    = blockIdx.x * (blockDim.x >> 5) + wave;
    const int eBase0 = job * (16 * TPW);
    if (eBase0 + 16 * TPW > nEdges) return;       // wave-uniform guard

    int   srcAll[TPW];
    float bfRow[TPW][16];                         // per-lane bond row, 4 tiles
    v2f   Anh[TPW][4];                            // neigh A-layout chunks

#pragma unroll
    for (int g = 0; g < TPW; ++g) {
        const long e  = (long)(eBase0 + g * 16 + e16);
        const int dst = pair[e * 2 + 1];
        srcAll[g]     = pair[e * 2 + 0];
#pragma unroll
        for (int q = 0; q < 4; ++q)
            *(v4f*)&bfRow[g][q * 4] = *(const v4f*)&bond[e * ATOM_DIM + q * 4];
#pragma unroll
        for (int q = 0; q < 4; ++q)
            Anh[g][q] = *(const v2f*)&atom[(long)dst * ATOM_DIM + 4 * q + 2 * half];
    }

    v8f acc[TPW];
#pragma unroll
    for (int g = 0; g < TPW; ++g)
        acc[g] = (v8f){0.f,0.f,0.f,0.f,0.f,0.f,0.f,0.f};

#pragma unroll
    for (int c0 = 0; c0 < NCHUNK; ++c0) {
        v2f B = *(const v2f*)&ldsW[((c0 * 2 + half) * 16 + e16) * 2];
#pragma unroll
        for (int g = 0; g < TPW; ++g) {
            // A-chunk of U: bf-scaled for kernel chunks, identity for bias.
            v2f A = (c0 < 64) ? (Anh[g][c0 & 3] * bfRow[g][c0 >> 2])
                              : Anh[g][c0 & 3];
            acc[g] = __builtin_amdgcn_wmma_f32_16x16x4_f32(
                         false, A, false, B, (short)0, acc[g], false, false);
        }
    }

    // Segment-sum scatter: element (e = v + 8*half, i = e16) of each tile.
#pragma unroll
    for (int g = 0; g < TPW; ++g) {
#pragma unroll
        for (int v = 0; v < 8; ++v) {
            int seg = __shfl(srcAll[g], ((lane & 16) >> 1) + v, 32);
            unsafeAtomicAdd(&out[(long)seg * ATOM_DIM + e16], acc[g][v]);
        }
    }
}

// ---------------------------------------------------------------------------
extern "C" void kernel_launch(void* const* d_in, const int* in_sizes, int n_in,
                              void* d_out, int out_size, void* d_ws, size_t ws_size,
                              hipStream_t stream) {
    const float* atom = (const float*)d_in[0];   // (20000,16) f32
    const float* bond = (const float*)d_in[1];   // (640000,16) f32
    const int*   pair = (const int*)d_in[2];     // (640000,2) i32
    const float* kern = (const float*)d_in[3];   // (16,256) f32
    const float* bias = (const float*)d_in[4];   // (256,) f32
    float*       out  = (float*)d_out;           // (20000,16) f32

    const int nEdges = in_sizes[1] / ATOM_DIM;

    zero_kernel<<<(out_size + 255) / 256, 256, 0, stream>>>(out, out_size);

    const int jobs   = nEdges / (16 * TPW);      // 10000 (E % 64 == 0)
    const int blocks = (jobs + 7) / 8;           // 8 waves (jobs) per block
    edge_net_kernel<<<blocks, 256, 0, stream>>>(atom, bond, pair, kern, bias,
                                                out, nEdges);
}